// Disentangle_Contrast_2199023255632
// MI455X (gfx1250) — compile-verified
//
#include <hip/hip_runtime.h>
#include <stdint.h>

#define B_    4
#define C_    128
#define HW_   16384
#define NC_   19
#define NP_   20
#define NPAD_ 32
#define INV_TEMP 14.2857142857142857f  // 1/0.07

typedef __attribute__((ext_vector_type(2))) float v2f;
typedef __attribute__((ext_vector_type(4))) float v4f;
typedef __attribute__((ext_vector_type(8))) float v8f;

// ---------------- meta zero ----------------
__global__ void k_zero(int* meta) {
    int i = threadIdx.x;
    if (i < 384) meta[i] = 0;   // counts[128] | cntPj[128] | cursor[128]
}

// ---------------- histograms ----------------
__global__ void k_count(const int* __restrict__ labels, const int* __restrict__ pj,
                        int* counts, int* cntPj) {
    int i = blockIdx.x * 256 + threadIdx.x;
    if (i >= B_ * HW_) return;
    int b = i / HW_;
    int lab = labels[i];
    if (lab >= 0 && lab < NC_) atomicAdd(&counts[b * NC_ + lab], 1);
    int c = pj[i];
    if (c >= 0 && c < NC_) atomicAdd(&cntPj[b * NC_ + c], 1);
}

// ---------------- exclusive scan over 76 entries ----------------
__global__ void k_scan(const int* __restrict__ cntPj, int* listOff) {
    if (blockIdx.x == 0 && threadIdx.x == 0) {
        int acc = 0;
        for (int i = 0; i < B_ * NC_; ++i) { listOff[i] = acc; acc += cntPj[i]; }
        listOff[B_ * NC_] = acc;
    }
}

// ---------------- scatter anchors into per-class lists ----------------
__global__ void k_build(const int* __restrict__ pj, const int* __restrict__ listOff,
                        int* cursor, int* list) {
    int i = blockIdx.x * 256 + threadIdx.x;
    if (i >= B_ * HW_) return;
    int b = i / HW_, q = i - b * HW_;
    int c = pj[i];
    if (c < 0 || c >= NC_) return;
    int slot = atomicAdd(&cursor[b * NC_ + c], 1);
    list[listOff[b * NC_ + c] + slot] = q;
}

// ---------------- top-20 of masked noise per (b,c) ----------------
// key = (order-preserving float bits << 32) | (0xFFFFFFFF - q): max-key == JAX top_k order
__global__ void k_topk(const float* __restrict__ noise, const int* __restrict__ labels,
                       int* neg_idx) {
    int g = blockIdx.x;              // b*NC + c
    int b = g / NC_, c = g - b * NC_;
    const float* nz = noise + (size_t)g * HW_;
    const int* lb = labels + b * HW_;
    __shared__ unsigned long long skey[256];
    unsigned long long prev = ~0ull;
    for (int r = 0; r < NP_; ++r) {
        unsigned long long best = 0;
        for (int q = threadIdx.x; q < HW_; q += 256) {
            float v = (lb[q] == c) ? nz[q] : -1e9f;
            unsigned int fb = __float_as_uint(v);
            fb = (fb & 0x80000000u) ? ~fb : (fb | 0x80000000u);
            unsigned long long key =
                ((unsigned long long)fb << 32) | (unsigned long long)(0xFFFFFFFFu - (unsigned)q);
            if (key < prev && key > best) best = key;
        }
        skey[threadIdx.x] = best;
        __syncthreads();
        for (int s = 128; s > 0; s >>= 1) {
            if ((int)threadIdx.x < s) {
                unsigned long long o = skey[threadIdx.x + s];
                if (o > skey[threadIdx.x]) skey[threadIdx.x] = o;
            }
            __syncthreads();
        }
        unsigned long long win = skey[0];
        if (threadIdx.x == 0)
            neg_idx[g * NP_ + r] = (int)(0xFFFFFFFFu - (unsigned)(win & 0xFFFFFFFFull));
        prev = win;
        __syncthreads();
    }
}

// ---------------- gather negatives: negTg[b][c][p(32)][k(128)] ----------------
__global__ void k_gather(const float* __restrict__ fq, const int* __restrict__ neg_idx,
                         float* negTg) {
    int g = blockIdx.x;              // b*NC + c
    int b = g / NC_;
    for (int idx = threadIdx.x; idx < NPAD_ * C_; idx += 256) {
        int p = idx >> 7, k = idx & (C_ - 1);
        float v = 0.f;
        if (p < NP_) {
            int q = neg_idx[g * NP_ + p];
            v = fq[((size_t)(b * C_ + k)) * HW_ + q];
        }
        negTg[(size_t)g * NPAD_ * C_ + idx] = v;
    }
}

// ---------------- transpose fq [B,C,HW] -> fqT [B,HW,C] ----------------
__global__ void k_transpose(const float* __restrict__ fq, float* __restrict__ fqT) {
    __shared__ float tile[32][33];
    int b = blockIdx.z;
    int q0 = blockIdx.x * 32, c0 = blockIdx.y * 32;
    int tx = threadIdx.x, ty = threadIdx.y;
    for (int j = ty; j < 32; j += 8)
        tile[j][tx] = fq[((size_t)(b * C_ + c0 + j)) * HW_ + q0 + tx];
    __syncthreads();
    for (int j = ty; j < 32; j += 8)
        fqT[((size_t)(b * HW_ + q0 + j)) * C_ + c0 + tx] = tile[tx][j];
}

// ---------------- l_pos[b,q] = sum_c fq*fk (coalesced) ----------------
__global__ void k_lpos(const float* __restrict__ fq, const float* __restrict__ fk,
                       float* lpos) {
    int i = blockIdx.x * 256 + threadIdx.x;
    if (i >= B_ * HW_) return;
    int b = i / HW_, q = i - b * HW_;
    const float* pq = fq + (size_t)b * C_ * HW_ + q;
    const float* pk = fk + (size_t)b * C_ * HW_ + q;
    float acc = 0.f;
#pragma unroll 4
    for (int c = 0; c < C_; ++c) acc += pq[(size_t)c * HW_] * pk[(size_t)c * HW_];
    lpos[i] = acc;
}

// ---------------- main: per-(b,c) skinny GEMM via f32 WMMA + CE epilogue ----------------
#define AST 132   // LDS row stride (floats), breaks 64-bank conflicts for strided lane access
__global__ void __launch_bounds__(32)
k_contrast(const float* __restrict__ fqT, const float* __restrict__ negTg,
           const int* __restrict__ list, const int* __restrict__ listOff,
           const int* __restrict__ cntPj, const int* __restrict__ counts,
           const int* __restrict__ predicts, const int* __restrict__ labels,
           const float* __restrict__ lpos, float* __restrict__ out) {
    __shared__ float negB[NPAD_][AST];   // [p][k]
    __shared__ float At[16][AST];        // [m][k]
    __shared__ float Dls[16][33];        // [m][p]

    int g = blockIdx.x;                  // b*NC + c
    int b = g / NC_, c = g - b * NC_;
    int nA = cntPj[g];
    int off = listOff[g];
    int tiles = (nA + 15) >> 4;
    if ((int)blockIdx.y >= tiles) return;       // uniform exit: EXEC stays all-ones

    int lane = threadIdx.x;
    int m16 = lane & 15, h = lane >> 4;

    // stage this class's 32x128 negative block (zero-padded cols 20..31)
    const float* nsrc = negTg + (size_t)g * NPAD_ * C_;
    for (int idx = lane; idx < NPAD_ * C_; idx += 32) {
        negB[idx >> 7][idx & (C_ - 1)] = nsrc[idx];
    }
    int cntL = counts[g];

    for (int t = blockIdx.y; t < tiles; t += gridDim.y) {
        __syncthreads();
        // stage 16 anchor feature rows (contiguous 512B each after transpose)
        for (int r = 0; r < 16; ++r) {
            int aidx = t * 16 + r;
            if (aidx > nA - 1) aidx = nA - 1;   // pad last tile with duplicate anchor
            int q = list[off + aidx];
            const float* row = fqT + ((size_t)(b * HW_ + q)) * C_;
            *(v4f*)&At[r][lane * 4] = *(const v4f*)(row + lane * 4);
        }
        __syncthreads();

        v8f acc0 = {}, acc1 = {};
#pragma unroll 8
        for (int k0 = 0; k0 < C_; k0 += 4) {
            int ko = k0 + 2 * h;
            v2f a  = *(const v2f*)&At[m16][ko];
            v2f b0 = *(const v2f*)&negB[m16][ko];
            v2f b1 = *(const v2f*)&negB[16 + m16][ko];
            acc0 = __builtin_amdgcn_wmma_f32_16x16x4_f32(false, a, false, b0,
                                                         (short)0, acc0, false, false);
            acc1 = __builtin_amdgcn_wmma_f32_16x16x4_f32(false, a, false, b1,
                                                         (short)0, acc1, false, false);
        }

        // dump D tile to LDS: lane L holds D[r + 8*(L/16)][L%16 (+16)]
#pragma unroll
        for (int r = 0; r < 8; ++r) {
            int m = r + 8 * h;
            Dls[m][m16]      = acc0[r];
            Dls[m][16 + m16] = acc1[r];
        }
        __syncthreads();

        if (lane < 16) {
            int aidx = t * 16 + lane;
            if (aidx < nA) {
                int q  = list[off + aidx];
                int gi = b * HW_ + q;
                float z0 = lpos[gi] * INV_TEMP;
                float mx = z0;
                float zn[NP_];
#pragma unroll
                for (int p = 0; p < NP_; ++p) {
                    zn[p] = Dls[lane][p] * INV_TEMP;
                    mx = fmaxf(mx, zn[p]);
                }
                float s = expf(z0 - mx);
#pragma unroll
                for (int p = 0; p < NP_; ++p) s += expf(zn[p] - mx);
                float loss = logf(s) + mx - z0;
                int pr = predicts[gi], la = labels[gi];
                bool valid = (pr != c) && (la != 255) && (pr == la) && (cntL >= NP_);
                out[gi] = valid ? loss : 0.0f;
            }
        }
    }
}

extern "C" void kernel_launch(void* const* d_in, const int* in_sizes, int n_in,
                              void* d_out, int out_size, void* d_ws, size_t ws_size,
                              hipStream_t stream) {
    const float* fq    = (const float*)d_in[0];
    const float* fk    = (const float*)d_in[1];
    const float* noise = (const float*)d_in[2];
    const int*   pr    = (const int*)d_in[3];
    const int*   pj    = (const int*)d_in[4];
    const int*   lab   = (const int*)d_in[5];
    float* out = (float*)d_out;

    char* ws = (char*)d_ws;
    size_t o = 0;
    auto alloc = [&](size_t bytes) { size_t r = o; o += (bytes + 255) & ~(size_t)255; return r; };

    int* meta    = (int*)(ws + alloc(512 * sizeof(int)));
    int* counts  = meta;         // label histogram   [B*NC]
    int* cntPj   = meta + 128;   // pj histogram      [B*NC]
    int* cursor  = meta + 256;   // scatter cursors   [B*NC]
    int* listOff = meta + 384;   // CSR offsets       [B*NC+1]
    int*   neg_idx = (int*)(ws + alloc((size_t)B_ * NC_ * NP_ * 4));
    int*   list    = (int*)(ws + alloc((size_t)B_ * HW_ * 4));
    float* lpos    = (float*)(ws + alloc((size_t)B_ * HW_ * 4));
    float* negTg   = (float*)(ws + alloc((size_t)B_ * NC_ * NPAD_ * C_ * 4));
    float* fqT     = (float*)(ws + alloc((size_t)B_ * HW_ * C_ * 4));
    (void)ws_size; (void)in_sizes; (void)n_in; (void)out_size;

    k_zero<<<1, 512, 0, stream>>>(meta);
    k_count<<<(B_ * HW_ + 255) / 256, 256, 0, stream>>>(lab, pj, counts, cntPj);
    k_scan<<<1, 32, 0, stream>>>(cntPj, listOff);
    k_build<<<(B_ * HW_ + 255) / 256, 256, 0, stream>>>(pj, listOff, cursor, list);
    k_topk<<<B_ * NC_, 256, 0, stream>>>(noise, lab, neg_idx);
    k_gather<<<B_ * NC_, 256, 0, stream>>>(fq, neg_idx, negTg);
    k_transpose<<<dim3(HW_ / 32, C_ / 32, B_), dim3(32, 8), 0, stream>>>(fq, fqT);
    k_lpos<<<(B_ * HW_ + 255) / 256, 256, 0, stream>>>(fq, fk, lpos);

    dim3 grid(B_ * NC_, 64);
    k_contrast<<<grid, 32, 0, stream>>>(fqT, negTg, list, listOff, cntPj, counts,
                                        pr, lab, lpos, out);
}